// GRUCell_24945170055197
// MI455X (gfx1250) — compile-verified
//
#include <hip/hip_runtime.h>
#include <math.h>

#define BATCHN 16
#define NODEN  2048
#define EMB    10
#define DP     12      // EMB padded to 3 K-chunks of 4
#define CDIM   66
#define CP     80      // CDIM padded to 5 col-frags of 16
#define HID    64
#define MH1    16
#define MH2    2

typedef float v2f __attribute__((ext_vector_type(2)));
typedef float v8f __attribute__((ext_vector_type(8)));

__device__ __forceinline__ float sigmoid_(float v) { return 1.0f / (1.0f + expf(-v)); }
// single-instruction relu: v_med3_f32(x, 0, FLT_MAX)
__device__ __forceinline__ float relu_(float v) {
    return __builtin_amdgcn_fmed3f(v, 0.0f, 3.402823466e+38f);
}
// low 32 bits of a generic pointer to LDS == DS address (flat->local is truncation)
__device__ __forceinline__ unsigned lds_lo32(const void* p) {
    return (unsigned)(unsigned long long)p;
}

// ---------------------------------------------------------------------------
// Kernel 1: per-(b,n) MLP -> nodevec (padded to DP), plus padded concat input.
// use_z==0: xs = [x, state]; use_z==1: xs = [x, z*state]
// ---------------------------------------------------------------------------
__global__ void __launch_bounds__(256) mlp_nodevec_kernel(
    const float* __restrict__ x, const float* __restrict__ state,
    const float* __restrict__ zbuf, int use_z,
    const float* __restrict__ ne0,
    const float* __restrict__ w1, const float* __restrict__ b1,
    const float* __restrict__ w2, const float* __restrict__ b2,
    const float* __restrict__ w3, const float* __restrict__ b3,
    float* __restrict__ xs_pad, float* __restrict__ nv)
{
    int idx = blockIdx.x * blockDim.x + threadIdx.x;   // b*N + n
    if (idx >= BATCHN * NODEN) return;
    int n = idx % NODEN;

    float xs[CDIM];
    xs[0] = x[idx * 2 + 0];
    xs[1] = x[idx * 2 + 1];
    const float* st = state + (size_t)idx * HID;
    if (use_z) {
        const float* z = zbuf + (size_t)idx * HID;
        #pragma unroll
        for (int i = 0; i < HID; ++i) xs[2 + i] = z[i] * st[i];
    } else {
        #pragma unroll
        for (int i = 0; i < HID; ++i) xs[2 + i] = st[i];
    }
    float* xo = xs_pad + (size_t)idx * CP;
    #pragma unroll
    for (int i = 0; i < CDIM; ++i) xo[i] = xs[i];
    #pragma unroll
    for (int i = CDIM; i < CP; ++i) xo[i] = 0.0f;

    float h1[MH1];
    #pragma unroll
    for (int h = 0; h < MH1; ++h) {
        float acc = b1[h];
        #pragma unroll
        for (int i = 0; i < CDIM; ++i) acc += xs[i] * w1[i * MH1 + h];
        h1[h] = sigmoid_(acc);
    }
    float h2[MH2];
    #pragma unroll
    for (int h = 0; h < MH2; ++h) {
        float acc = b2[h];
        #pragma unroll
        for (int i = 0; i < MH1; ++i) acc += h1[i] * w2[i * MH2 + h];
        h2[h] = sigmoid_(acc);
    }
    float* nvo = nv + (size_t)idx * DP;
    #pragma unroll
    for (int d = 0; d < EMB; ++d) {
        float acc = b3[d];
        #pragma unroll
        for (int i = 0; i < MH2; ++i) acc += h2[i] * w3[i * EMB + d];
        nvo[d] = tanhf(ne0[n * EMB + d] * acc);
    }
    nvo[10] = 0.0f; nvo[11] = 0.0f;
}

// ---------------------------------------------------------------------------
// Kernel 2: d[b,n] = rsqrt( sum_m relu(nv[b,n,:] . nv[b,m,:]) )
// One wave = 16 rows; 8 waves/WG = 128 rows. G tiles via WMMA f32 16x16x4.
// ---------------------------------------------------------------------------
__global__ void __launch_bounds__(256) rowsum_kernel(
    const float* __restrict__ nv, float* __restrict__ dvec)
{
    int b     = blockIdx.y;
    int wave  = threadIdx.x >> 5;
    int lane  = threadIdx.x & 31;
    int lrow  = lane & 15;
    int hi    = lane >> 4;
    int khalf = hi ? 2 : 0;
    int n0    = blockIdx.x * 128 + wave * 16;
    const float* nvb = nv + (size_t)b * NODEN * DP;

    v2f afr[3];
    const float* arow = nvb + (size_t)(n0 + lrow) * DP;
    #pragma unroll
    for (int kc = 0; kc < 3; ++kc) {
        afr[kc][0] = arow[kc * 4 + khalf];
        afr[kc][1] = arow[kc * 4 + khalf + 1];
    }

    v8f acc = {};
    for (int m0 = 0; m0 < NODEN; m0 += 16) {
        const float* brow = nvb + (size_t)(m0 + lrow) * DP;   // column m0+lrow
        v8f c = {};
        #pragma unroll
        for (int kc = 0; kc < 3; ++kc) {
            v2f bfr;
            bfr[0] = brow[kc * 4 + khalf];
            bfr[1] = brow[kc * 4 + khalf + 1];
            c = __builtin_amdgcn_wmma_f32_16x16x4_f32(false, afr[kc], false, bfr,
                                                      (short)0, c, false, false);
        }
        #pragma unroll
        for (int v = 0; v < 8; ++v) acc[v] += relu_(c[v]);
    }
    // reduce each half-wave's 16 lanes (C/D layout: lane = column index)
    #pragma unroll
    for (int v = 0; v < 8; ++v) {
        float s = acc[v];
        s += __shfl_xor(s, 1, 32);
        s += __shfl_xor(s, 2, 32);
        s += __shfl_xor(s, 4, 32);
        s += __shfl_xor(s, 8, 32);
        if (lrow == 0) dvec[(size_t)b * NODEN + n0 + v + (hi ? 8 : 0)] = rsqrtf(s);
    }
}

// ---------------------------------------------------------------------------
// Kernel 3: y2 = D * relu(nv nv^T) * D @ xs.
// Double-buffered GLOBAL_LOAD_ASYNC_TO_LDS_B128 streams the X/nv/d m-tiles
// into LDS (ASYNCcnt), overlapped with the WMMA pipeline.
// ---------------------------------------------------------------------------
__global__ void __launch_bounds__(256) graphconv_kernel(
    const float* __restrict__ nv, const float* __restrict__ dvec,
    const float* __restrict__ xs, float* __restrict__ y2)
{
    __shared__ __align__(16) float sX[2][16][84];   // 84-stride: conflict-free halves
    __shared__ __align__(16) float sNV[2][16][12];
    __shared__ __align__(16) float sD[2][16];
    __shared__ float sG[8][16][20];                 // per-wave G tile, stride 20

    int b     = blockIdx.y;
    int wave  = threadIdx.x >> 5;
    int lane  = threadIdx.x & 31;
    int lrow  = lane & 15;
    int hi    = lane >> 4;
    int khalf = hi ? 2 : 0;
    int n0    = blockIdx.x * 128 + wave * 16;
    const float* nvb = nv + (size_t)b * NODEN * DP;
    const float* xsb = xs + (size_t)b * NODEN * CP;
    const float* db  = dvec + (size_t)b * NODEN;

    // --- per-thread async chunk descriptors: 2 x 16B chunks per tile -------
    // chunk ids: [0,320) X tile, [320,368) nv tile, [368,372) d tile,
    // ids >= 372 wrap (benign duplicate transfers keep ASYNCcnt uniform).
    unsigned long long gcur[2];
    unsigned gstep[2], lbase[2], lstep[2];
    #pragma unroll
    for (int q = 0; q < 2; ++q) {
        int id = (int)threadIdx.x + q * 256;
        if (id >= 372) id -= 372;
        if (id < 320) {
            int r = id / 20, c = id % 20;
            gcur[q]  = (unsigned long long)(const void*)(xsb + r * CP + c * 4);
            gstep[q] = 16 * CP * 4;
            lbase[q] = lds_lo32(&sX[0][r][c * 4]);
            lstep[q] = lds_lo32(&sX[1][0][0]) - lds_lo32(&sX[0][0][0]);
        } else if (id < 368) {
            int id2 = id - 320;
            int r = id2 / 3, c = id2 % 3;
            gcur[q]  = (unsigned long long)(const void*)(nvb + r * DP + c * 4);
            gstep[q] = 16 * DP * 4;
            lbase[q] = lds_lo32(&sNV[0][r][c * 4]);
            lstep[q] = lds_lo32(&sNV[1][0][0]) - lds_lo32(&sNV[0][0][0]);
        } else {
            int c = id - 368;
            gcur[q]  = (unsigned long long)(const void*)(db + c * 4);
            gstep[q] = 16 * 4;
            lbase[q] = lds_lo32(&sD[0][c * 4]);
            lstep[q] = lds_lo32(&sD[1][0]) - lds_lo32(&sD[0][0]);
        }
    }
    auto issue = [&](int bufsel) {
        #pragma unroll
        for (int q = 0; q < 2; ++q) {
            asm volatile("global_load_async_to_lds_b128 %0, %1, off"
                         :: "v"(lbase[q] + (unsigned)bufsel * lstep[q]),
                            "v"(gcur[q])
                         : "memory");
            gcur[q] += gstep[q];
        }
    };

    v2f afr[3];
    const float* arow = nvb + (size_t)(n0 + lrow) * DP;
    #pragma unroll
    for (int kc = 0; kc < 3; ++kc) {
        afr[kc][0] = arow[kc * 4 + khalf];
        afr[kc][1] = arow[kc * 4 + khalf + 1];
    }

    v8f out[5] = {};

    issue(0);                                    // prefetch tile 0
    const int NT = NODEN / 16;
    for (int t = 0; t < NT; ++t) {
        int buf = t & 1;
        if (t + 1 < NT) {
            issue(buf ^ 1);                      // prefetch next tile
            asm volatile("s_wait_asynccnt 0x2" ::: "memory");  // tile t landed
        } else {
            asm volatile("s_wait_asynccnt 0x0" ::: "memory");
        }
        __syncthreads();                         // all waves' transfers visible

        // G tile = nv_rows @ nv_cols^T  (K = 12)
        v8f g = {};
        #pragma unroll
        for (int kc = 0; kc < 3; ++kc) {
            v2f bfr;
            bfr[0] = sNV[buf][lrow][kc * 4 + khalf];
            bfr[1] = sNV[buf][lrow][kc * 4 + khalf + 1];
            g = __builtin_amdgcn_wmma_f32_16x16x4_f32(false, afr[kc], false, bfr,
                                                      (short)0, g, false, false);
        }
        float dm = sD[buf][lrow];                // lane = column m in C/D layout
        #pragma unroll
        for (int v = 0; v < 8; ++v) g[v] = relu_(g[v]) * dm;

        // stage G (C/D layout -> row-major LDS), reload as A frags.
        // Same-wave LDS store->load is in-order per ISA; block compiler reordering.
        #pragma unroll
        for (int v = 0; v < 8; ++v) sG[wave][v + (hi ? 8 : 0)][lrow] = g[v];
        asm volatile("" ::: "memory");

        #pragma unroll
        for (int kc = 0; kc < 4; ++kc) {
            v2f ga;
            ga[0] = sG[wave][lrow][kc * 4 + khalf];
            ga[1] = sG[wave][lrow][kc * 4 + khalf + 1];
            #pragma unroll
            for (int f = 0; f < 5; ++f) {
                v2f bx;
                bx[0] = sX[buf][kc * 4 + khalf][f * 16 + lrow];
                bx[1] = sX[buf][kc * 4 + khalf + 1][f * 16 + lrow];
                out[f] = __builtin_amdgcn_wmma_f32_16x16x4_f32(false, ga, false, bx,
                                                               (short)0, out[f], false, false);
            }
        }
        __syncthreads();                         // done reading buf this iter
    }

    float* y2b = y2 + (size_t)b * NODEN * CP;
    #pragma unroll
    for (int v = 0; v < 8; ++v) {
        int r = v + (hi ? 8 : 0);
        float dn = db[n0 + r];
        #pragma unroll
        for (int f = 0; f < 5; ++f)
            y2b[(size_t)(n0 + r) * CP + f * 16 + lrow] = out[f][v] * dn;
    }
}

// ---------------------------------------------------------------------------
// Kernel 4: per-node output mix. One WG per node.
//   out[b,o] = xs[b,:] @ W0 + y2[b,:] @ W1 + bias,  W_k = sum_d ne1[n,d] wp[d,k]
// MODE 0 (gate, O=128): z = sigmoid(out[:, :64]), r = sigmoid(out[:, 64:])
// MODE 1 (upd,  O=64) : hc = tanh(out); h = r*state + (1-r)*hc -> d_out
// ---------------------------------------------------------------------------
template <int O, int MODE>
__global__ void __launch_bounds__(256) nodemix_kernel(
    const float* __restrict__ xs, const float* __restrict__ y2,
    const float* __restrict__ ne1,
    const float* __restrict__ wp, const float* __restrict__ bp,
    float* __restrict__ out0, float* __restrict__ out1,
    const float* __restrict__ rbuf, const float* __restrict__ state)
{
    __shared__ float sW[CP][O + 8];        // one k-slice of per-node weights
    __shared__ float sA[2][16][CP + 4];    // [0]=xs tile (batch x CP), [1]=y2 tile
    __shared__ float sBias[O];
    __shared__ float sNe[EMB];

    int n     = blockIdx.x;
    int tid   = threadIdx.x;
    int wave  = tid >> 5;
    int lane  = tid & 31;
    int lrow  = lane & 15;
    int hi    = lane >> 4;
    int khalf = hi ? 2 : 0;
    const int cf = wave;                   // col-frag owned by this wave
    const bool active = (wave < O / 16);

    if (tid < EMB) sNe[tid] = ne1[n * EMB + tid];
    for (int i = tid; i < 16 * CP; i += 256) {
        int bb = i / CP, c = i % CP;
        size_t base = ((size_t)bb * NODEN + n) * CP + c;
        sA[0][bb][c] = xs[base];
        sA[1][bb][c] = y2[base];
    }
    __syncthreads();

    for (int o = tid; o < O; o += 256) {
        float acc = 0.0f;
        #pragma unroll
        for (int d = 0; d < EMB; ++d) acc += sNe[d] * bp[d * O + o];
        sBias[o] = acc;
    }

    v8f cfr = {};
    for (int kk = 0; kk < 2; ++kk) {
        __syncthreads();                   // sW safe to overwrite
        for (int i = tid; i < CP * O; i += 256) {
            int ii = i / O, oo = i % O;
            float acc = 0.0f;
            if (ii < CDIM) {
                #pragma unroll
                for (int d = 0; d < EMB; ++d)
                    acc += sNe[d] * wp[((size_t)(d * 2 + kk) * CDIM + ii) * O + oo];
            }
            sW[ii][oo] = acc;
        }
        __syncthreads();
        if (active) {                      // wave-uniform -> EXEC stays all-ones
            #pragma unroll
            for (int kc = 0; kc < 20; ++kc) {
                v2f a;
                a[0] = sA[kk][lrow][kc * 4 + khalf];
                a[1] = sA[kk][lrow][kc * 4 + khalf + 1];
                v2f bw;
                bw[0] = sW[kc * 4 + khalf][cf * 16 + lrow];
                bw[1] = sW[kc * 4 + khalf + 1][cf * 16 + lrow];
                cfr = __builtin_amdgcn_wmma_f32_16x16x4_f32(false, a, false, bw,
                                                            (short)0, cfr, false, false);
            }
        }
    }

    if (active) {
        #pragma unroll
        for (int v = 0; v < 8; ++v) {
            int bb = v + (hi ? 8 : 0);                 // batch index
            int o  = cf * 16 + lrow;
            float val = cfr[v] + sBias[o];
            if (MODE == 0) {
                float s = sigmoid_(val);
                if (cf < 4) out0[((size_t)bb * NODEN + n) * HID + o] = s;
                else        out1[((size_t)bb * NODEN + n) * HID + (o - HID)] = s;
            } else {
                size_t idx = ((size_t)bb * NODEN + n) * HID + o;
                float hc = tanhf(val);
                float r  = rbuf[idx];
                out0[idx] = r * state[idx] + (1.0f - r) * hc;
            }
        }
    }
}

// ---------------------------------------------------------------------------
extern "C" void kernel_launch(void* const* d_in, const int* in_sizes, int n_in,
                              void* d_out, int out_size, void* d_ws, size_t ws_size,
                              hipStream_t stream)
{
    const float* x     = (const float*)d_in[0];
    const float* state = (const float*)d_in[1];
    const float* ne0   = (const float*)d_in[2];
    const float* ne1   = (const float*)d_in[3];
    const float* g_wp = (const float*)d_in[4];  const float* g_bp = (const float*)d_in[5];
    const float* g_w1 = (const float*)d_in[6];  const float* g_b1 = (const float*)d_in[7];
    const float* g_w2 = (const float*)d_in[8];  const float* g_b2 = (const float*)d_in[9];
    const float* g_w3 = (const float*)d_in[10]; const float* g_b3 = (const float*)d_in[11];
    const float* u_wp = (const float*)d_in[12]; const float* u_bp = (const float*)d_in[13];
    const float* u_w1 = (const float*)d_in[14]; const float* u_b1 = (const float*)d_in[15];
    const float* u_w2 = (const float*)d_in[16]; const float* u_b2 = (const float*)d_in[17];
    const float* u_w3 = (const float*)d_in[18]; const float* u_b3 = (const float*)d_in[19];
    float* out = (float*)d_out;

    float* p = (float*)d_ws;
    float* xs_pad   = p; p += (size_t)BATCHN * NODEN * CP;
    float* cand_pad = p; p += (size_t)BATCHN * NODEN * CP;
    float* nv       = p; p += (size_t)BATCHN * NODEN * DP;
    float* dvec     = p; p += (size_t)BATCHN * NODEN;
    float* y2       = p; p += (size_t)BATCHN * NODEN * CP;
    float* zbuf     = p; p += (size_t)BATCHN * NODEN * HID;
    float* rbuf     = p; p += (size_t)BATCHN * NODEN * HID;

    dim3 blk(256);
    dim3 gridA((BATCHN * NODEN) / 256);
    dim3 gridG(NODEN / 128, BATCHN);
    dim3 gridN(NODEN);

    // ---- gate GCN ----
    mlp_nodevec_kernel<<<gridA, blk, 0, stream>>>(x, state, nullptr, 0, ne0,
        g_w1, g_b1, g_w2, g_b2, g_w3, g_b3, xs_pad, nv);
    rowsum_kernel<<<gridG, blk, 0, stream>>>(nv, dvec);
    graphconv_kernel<<<gridG, blk, 0, stream>>>(nv, dvec, xs_pad, y2);
    nodemix_kernel<128, 0><<<gridN, blk, 0, stream>>>(xs_pad, y2, ne1, g_wp, g_bp,
        zbuf, rbuf, nullptr, nullptr);

    // ---- update GCN + GRU blend ----
    mlp_nodevec_kernel<<<gridA, blk, 0, stream>>>(x, state, zbuf, 1, ne0,
        u_w1, u_b1, u_w2, u_b2, u_w3, u_b3, cand_pad, nv);
    rowsum_kernel<<<gridG, blk, 0, stream>>>(nv, dvec);
    graphconv_kernel<<<gridG, blk, 0, stream>>>(nv, dvec, cand_pad, y2);
    nodemix_kernel<64, 1><<<gridN, blk, 0, stream>>>(cand_pad, y2, ne1, u_wp, u_bp,
        out, nullptr, rbuf, state);

    (void)in_sizes; (void)n_in; (void)out_size; (void)ws_size;
}